// GCN_35433480192701
// MI455X (gfx1250) — compile-verified
//
#include <hip/hip_runtime.h>
#include <hip/hip_bf16.h>

typedef __attribute__((ext_vector_type(2))) float v2f;
typedef __attribute__((ext_vector_type(8))) float v8f;

static __device__ __forceinline__ v8f wmma4(v2f a, v2f b, v8f c) {
    // 8 args: (neg_a, A, neg_b, B, c_mod, C, reuse_a, reuse_b)
    return __builtin_amdgcn_wmma_f32_16x16x4_f32(false, a, false, b, (short)0, c,
                                                 false, false);
}

// ---------------------------------------------------------------------------
// Degree / normalization
// ---------------------------------------------------------------------------
__global__ void GCN_deg_init(float* deg, int n) {
    int i = blockIdx.x * blockDim.x + threadIdx.x;
    if (i < n) deg[i] = 1.0f;               // self-loop
}

__global__ void GCN_deg_count(const int* __restrict__ dst, float* deg, int e) {
    int i = blockIdx.x * blockDim.x + threadIdx.x;
    if (i < e) atomicAdd(&deg[dst[i]], 1.0f);
}

__global__ void GCN_deg_to_dinv(float* deg, int n) {
    int i = blockIdx.x * blockDim.x + threadIdx.x;
    if (i < n) deg[i] = rsqrtf(deg[i]);
}

// ---------------------------------------------------------------------------
// Pack weight W[K,N] (row-major) into transposed, zero-padded Bt[Npad,Kpad]
// so that the WMMA B-fragment pair (k, k+1) for a fixed column is contiguous.
// ---------------------------------------------------------------------------
__global__ void GCN_pack_weight(const float* __restrict__ W, float* __restrict__ Bt,
                                int K, int N, int Kpad, int Npad) {
    int idx = blockIdx.x * blockDim.x + threadIdx.x;
    if (idx >= Npad * Kpad) return;
    int n = idx / Kpad;
    int k = idx - n * Kpad;
    Bt[idx] = (k < K && n < N) ? W[k * N + n] : 0.0f;
}

// Optional zero-pad copy of a dense activation matrix (only used if the input
// feature count is not a multiple of 16; FIN=128 skips this).
__global__ void GCN_pad_rows(const float* __restrict__ X, float* __restrict__ Xp,
                             int n, int F, int Fpad) {
    int idx = blockIdx.x * blockDim.x + threadIdx.x;
    if (idx >= n * Fpad) return;
    int i = idx / Fpad;
    int c = idx - i * Fpad;
    Xp[idx] = (c < F) ? X[(long long)i * F + c] : 0.0f;
}

// ---------------------------------------------------------------------------
// GEMM via native fp32 WMMA: C[M, ldc] = A[M, lda(Kpad cols)] * Bt^T
//   - Kpad % 16 == 0, ldc == Npad (multiple of 16): inner loop is branch-free
//   - A row clamped for ragged last M tile (those rows are never stored)
//   - pad columns of C are written as 0 so downstream K-padding stays exact
// One wave32 per 16x16 tile; 4 WMMAs + 8 b64 loads per 16 k-steps.
// ---------------------------------------------------------------------------
__global__ void GCN_wmma_gemm_f32(const float* __restrict__ A, int lda,
                                  const float* __restrict__ Bt,
                                  float* __restrict__ C, int ldc,
                                  int M, int Kpad, int N) {
    const int lane = threadIdx.x & 31;
    const int wave = threadIdx.x >> 5;
    const int tilesN = ldc >> 4;
    const int tilesM = (M + 15) >> 4;
    const int tile = blockIdx.x * (blockDim.x >> 5) + wave;
    if (tile >= tilesM * tilesN) return;      // wave-uniform: EXEC all-ones below

    const int tm = (tile / tilesN) << 4;
    const int tn = (tile % tilesN) << 4;
    const int half = lane >> 4;               // 0: lanes 0-15, 1: lanes 16-31
    const int l16  = lane & 15;

    int mA = tm + l16;
    if (mA >= M) mA = M - 1;                  // clamp: affects only unstored rows
    const int nB = tn + l16;                  // < Npad by construction

    const float* ap = A  + (long long)mA * lda  + (half << 1);
    const float* bp = Bt + (long long)nB * Kpad + (half << 1);

    v8f acc = {};
    for (int k0 = 0; k0 < Kpad; k0 += 16) {
        v2f a0 = *(const v2f*)(ap + k0);
        v2f b0 = *(const v2f*)(bp + k0);
        v2f a1 = *(const v2f*)(ap + k0 + 4);
        v2f b1 = *(const v2f*)(bp + k0 + 4);
        v2f a2 = *(const v2f*)(ap + k0 + 8);
        v2f b2 = *(const v2f*)(bp + k0 + 8);
        v2f a3 = *(const v2f*)(ap + k0 + 12);
        v2f b3 = *(const v2f*)(bp + k0 + 12);
        acc = wmma4(a0, b0, acc);
        acc = wmma4(a1, b1, acc);
        acc = wmma4(a2, b2, acc);
        acc = wmma4(a3, b3, acc);
    }

    // C/D layout: VGPR v -> row tm+v (lanes 0-15) / tm+v+8 (lanes 16-31)
    const int n = tn + l16;
    const bool ncol = (n < N);
#pragma unroll
    for (int v = 0; v < 8; ++v) {
        const int m = tm + v + (half << 3);
        if (m < M) C[(long long)m * ldc + n] = ncol ? acc[v] : 0.0f;
    }
}

// ---------------------------------------------------------------------------
// agg[i, c<F] = h[i, c]*dinv[i]^2 + bias[c] ; agg[i, F<=c<lda] = 0
// ---------------------------------------------------------------------------
__global__ void GCN_agg_init(const float* __restrict__ h, int ldh,
                             const float* __restrict__ dinv,
                             const float* __restrict__ bias,
                             float* __restrict__ agg, int lda, int n, int f) {
    int idx = blockIdx.x * blockDim.x + threadIdx.x;
    if (idx >= n * lda) return;
    int i = idx / lda;
    int c = idx - i * lda;
    float d = dinv[i];
    agg[idx] = (c < f) ? h[(long long)i * ldh + c] * d * d + bias[c] : 0.0f;
}

// ---------------------------------------------------------------------------
// Edge scatter: agg[dst, c] += h[src, c] * dinv[src]*dinv[dst]
// One thread per (edge, 4-feature chunk): float4 gather (16B aligned via
// padded ldh), dinv loaded once per chunk, 4 scalar L2 atomics.
// ---------------------------------------------------------------------------
__global__ void GCN_scatter4(const float* __restrict__ h, int ldh,
                             const float* __restrict__ dinv,
                             const int* __restrict__ src,
                             const int* __restrict__ dst,
                             float* __restrict__ agg, int lda, int e, int f) {
    const int chunks = (f + 3) >> 2;
    int idx = blockIdx.x * blockDim.x + threadIdx.x;
    if (idx >= e * chunks) return;
    int ed = idx / chunks;
    int c4 = (idx - ed * chunks) << 2;
    int s = src[ed];
    int d = dst[ed];
    float norm = dinv[s] * dinv[d];
    float4 hv = *(const float4*)(h + (long long)s * ldh + c4);
    float* ap = agg + (long long)d * lda + c4;
    if (c4 + 0 < f) atomicAdd(ap + 0, hv.x * norm);
    if (c4 + 1 < f) atomicAdd(ap + 1, hv.y * norm);
    if (c4 + 2 < f) atomicAdd(ap + 2, hv.z * norm);
    if (c4 + 3 < f) atomicAdd(ap + 3, hv.w * norm);
}

// ---------------------------------------------------------------------------
// Activation: 0 = tanh, 1 = relu (pad columns hold 0; both map 0 -> 0)
// ---------------------------------------------------------------------------
__global__ void GCN_act(float* buf, int n, int mode) {
    int i = blockIdx.x * blockDim.x + threadIdx.x;
    if (i >= n) return;
    float v = buf[i];
    buf[i] = (mode == 0) ? tanhf(v) : fmaxf(v, 0.0f);
}

// ---------------------------------------------------------------------------
// Host orchestration
// ---------------------------------------------------------------------------
static inline int cdiv(long long a, int b) { return (int)((a + b - 1) / b); }
static inline int ceil16(int v) { return (v + 15) & ~15; }

static void run_gemm(const float* A, int lda, const float* Bt, float* C, int ldc,
                     int M, int Kpad, int N, hipStream_t s) {
    int tiles = cdiv(M, 16) * (ldc >> 4);
    GCN_wmma_gemm_f32<<<cdiv(tiles, 8), 256, 0, s>>>(A, lda, Bt, C, ldc, M, Kpad, N);
}

extern "C" void kernel_launch(void* const* d_in, const int* in_sizes, int n_in,
                              void* d_out, int out_size, void* d_ws, size_t ws_size,
                              hipStream_t stream) {
    const float* x  = (const float*)d_in[0];
    const int*   ei = (const int*)d_in[1];
    const float* W1 = (const float*)d_in[2];
    const float* b1 = (const float*)d_in[3];
    const float* W2 = (const float*)d_in[4];
    const float* b2 = (const float*)d_in[5];
    const float* W3 = (const float*)d_in[6];
    const float* b3 = (const float*)d_in[7];
    const float* W4 = (const float*)d_in[8];
    const float* b4 = (const float*)d_in[9];
    float* out = (float*)d_out;

    const int H   = in_sizes[3];
    const int D3  = in_sizes[5];
    const int D4  = in_sizes[7];
    const int C   = in_sizes[9];
    const int FIN = in_sizes[2] / H;
    const int N   = in_sizes[0] / FIN;
    const int E   = in_sizes[1] / 2;
    const int* src = ei;
    const int* dst = ei + E;

    // Padded dims (multiples of 16) for branch-free WMMA
    const int KP1 = ceil16(FIN);   // 128
    const int HP  = ceil16(H);     // 64
    const int D3P = ceil16(D3);    // 112
    const int D4P = ceil16(D4);    // 32
    const int CP  = ceil16(C);     // 32
    const int FP  = (HP > D3P ? HP : D3P);  // widest padded layer

    // Workspace layout
    float* dinv = (float*)d_ws;
    float* hbuf = dinv + N;
    float* aggA = hbuf + (long long)N * FP;
    float* aggB = aggA + (long long)N * FP;
    float* bt1  = aggB + (long long)N * FP;
    float* bt2  = bt1 + (long long)KP1 * HP;
    float* bt3  = bt2 + (long long)HP * D3P;
    float* bt4  = bt3 + (long long)D3P * D4P;
    float* xpad = bt4 + (long long)D4P * CP;  // only touched if FIN % 16 != 0

    const int T = 256;

    // ---- degree -> dinv ----
    GCN_deg_init   <<<cdiv(N, T), T, 0, stream>>>(dinv, N);
    GCN_deg_count  <<<cdiv(E, T), T, 0, stream>>>(dst, dinv, E);
    GCN_deg_to_dinv<<<cdiv(N, T), T, 0, stream>>>(dinv, N);

    // ---- pack weights (transposed + zero-padded; tiny) ----
    GCN_pack_weight<<<cdiv((long long)HP  * KP1, T), T, 0, stream>>>(W1, bt1, FIN, H,  KP1, HP);
    GCN_pack_weight<<<cdiv((long long)D3P * HP,  T), T, 0, stream>>>(W2, bt2, H,   D3, HP,  D3P);
    GCN_pack_weight<<<cdiv((long long)D4P * D3P, T), T, 0, stream>>>(W3, bt3, D3,  D4, D3P, D4P);
    GCN_pack_weight<<<cdiv((long long)CP  * D4P, T), T, 0, stream>>>(W4, bt4, D4,  C,  D4P, CP);

    // ---- input (pad only if needed; FIN=128 uses x directly) ----
    const float* A1 = x;
    if (FIN != KP1) {
        GCN_pad_rows<<<cdiv((long long)N * KP1, T), T, 0, stream>>>(x, xpad, N, FIN, KP1);
        A1 = xpad;
    }

    // ---- layer 1: tanh(GCNConv(x, W1)) ----
    run_gemm(A1, KP1, bt1, hbuf, HP, N, KP1, H, stream);
    GCN_agg_init<<<cdiv((long long)N * HP, T), T, 0, stream>>>(hbuf, HP, dinv, b1, aggA, HP, N, H);
    GCN_scatter4<<<cdiv((long long)E * ((H + 3) >> 2), T), T, 0, stream>>>(
        hbuf, HP, dinv, src, dst, aggA, HP, E, H);
    GCN_act<<<cdiv((long long)N * HP, T), T, 0, stream>>>(aggA, N * HP, 0);

    // ---- layer 2: relu ----
    run_gemm(aggA, HP, bt2, hbuf, D3P, N, HP, D3, stream);
    GCN_agg_init<<<cdiv((long long)N * D3P, T), T, 0, stream>>>(hbuf, D3P, dinv, b2, aggB, D3P, N, D3);
    GCN_scatter4<<<cdiv((long long)E * ((D3 + 3) >> 2), T), T, 0, stream>>>(
        hbuf, D3P, dinv, src, dst, aggB, D3P, E, D3);
    GCN_act<<<cdiv((long long)N * D3P, T), T, 0, stream>>>(aggB, N * D3P, 1);

    // ---- layer 3: relu ----
    run_gemm(aggB, D3P, bt3, hbuf, D4P, N, D3P, D4, stream);
    GCN_agg_init<<<cdiv((long long)N * D4P, T), T, 0, stream>>>(hbuf, D4P, dinv, b3, aggA, D4P, N, D4);
    GCN_scatter4<<<cdiv((long long)E * ((D4 + 3) >> 2), T), T, 0, stream>>>(
        hbuf, D4P, dinv, src, dst, aggA, D4P, E, D4);
    GCN_act<<<cdiv((long long)N * D4P, T), T, 0, stream>>>(aggA, N * D4P, 1);

    // ---- layer 4: dense output straight into d_out (no padding there) ----
    run_gemm(aggA, D4P, bt4, hbuf, CP, N, D4P, C, stream);
    GCN_agg_init<<<cdiv((long long)N * C, T), T, 0, stream>>>(hbuf, CP, dinv, b4, out, C, N, C);
    GCN_scatter4<<<cdiv((long long)E * ((C + 3) >> 2), T), T, 0, stream>>>(
        hbuf, CP, dinv, src, dst, out, C, E, C);
}